// GraphSage_21981642621367
// MI455X (gfx1250) — compile-verified
//
#include <hip/hip_runtime.h>

typedef __attribute__((ext_vector_type(2))) float v2f;
typedef __attribute__((ext_vector_type(8))) float v8f;

#define DIM    128
#define TWOD   256
#define TILE_M 16
#define HSTR   260   // padded LDS stride for h  (conflict-free: (r*260+c)%64 distinct over 16 rows)
#define WSTR   260   // padded LDS stride for W
#define ZSTR   132   // padded LDS stride for z tile

__global__ __launch_bounds__(256)
void graphsage_fused(const int* __restrict__ batch,
                     const int* __restrict__ neigh,
                     const float* __restrict__ Z,
                     const float* __restrict__ W,
                     float* __restrict__ out,
                     int B, int S, int nTiles)
{
    __shared__ float Wl[DIM * WSTR];      // 130.0 KB  staged W, padded
    __shared__ float hl[TILE_M * HSTR];   //  16.3 KB  h = [self | mean(neigh)]
    __shared__ float zt[TILE_M * ZSTR];   //   8.3 KB  sigmoid(z) tile
    __shared__ float part[TILE_M][16];    //   1.0 KB  norm partials
    __shared__ float normv[TILE_M];

    const int t    = threadIdx.x;
    const int wave = t >> 5;
    const int lane = t & 31;

    // ---- Stage W (128 x 256 row-major) into LDS once per block ----
    {
        const float4* Wv = reinterpret_cast<const float4*>(W);
        for (int i = t; i < (DIM * TWOD) / 4; i += 256) {
            int row = i >> 6;              // 64 float4 per row
            int c4  = i & 63;
            float4 v = Wv[i];
            *reinterpret_cast<float4*>(&Wl[row * WSTR + c4 * 4]) = v;
        }
    }
    __syncthreads();

    const float invS = 1.0f / (float)S;

    for (int tile = blockIdx.x; tile < nTiles; tile += gridDim.x) {
        const int row0 = tile * TILE_M;

        // ---- Phase 1: gather + mean aggregate -> hl[16][256] ----
        // wave w handles rows 2w, 2w+1; each lane owns 4 contiguous floats of a row
        for (int rr = 0; rr < 2; ++rr) {
            const int r = wave * 2 + rr;
            int gr = row0 + r;
            if (gr >= B) gr = B - 1;       // clamp (B is a multiple of 16 in practice)
            const int bi = batch[gr];
            const float4 self =
                *reinterpret_cast<const float4*>(Z + (size_t)bi * DIM + lane * 4);
            float4 acc = make_float4(0.f, 0.f, 0.f, 0.f);
            const int* nrow = neigh + (size_t)gr * S;
            for (int s = 0; s < S; ++s) {
                const int ni = nrow[s];
                const float4 nv =
                    *reinterpret_cast<const float4*>(Z + (size_t)ni * DIM + lane * 4);
                acc.x += nv.x; acc.y += nv.y; acc.z += nv.z; acc.w += nv.w;
            }
            acc.x *= invS; acc.y *= invS; acc.z *= invS; acc.w *= invS;
            *reinterpret_cast<float4*>(&hl[r * HSTR + lane * 4])       = self;
            *reinterpret_cast<float4*>(&hl[r * HSTR + DIM + lane * 4]) = acc;
        }
        __syncthreads();   // B1

        // ---- Phase 2: WMMA f32 GEMM: z[16, n0:n0+16] = h[16,256] @ W.T[256, n0:n0+16]
        {
            const int n0   = wave * 16;        // this wave's output-column tile
            const int sub  = lane & 15;        // A: M index ; B: (n - n0)
            const int koff = (lane >> 4) * 2;  // K pair selector per ISA 16x4 layout
            const float* ha = hl + sub * HSTR + koff;
            const float* wb = Wl + (n0 + sub) * WSTR + koff;  // B[kk][nn] = W[n0+nn][k+kk]

            v8f acc = {0.f,0.f,0.f,0.f,0.f,0.f,0.f,0.f};
            #pragma unroll
            for (int k = 0; k < TWOD; k += 4) {
                v2f a, b;
                a.x = ha[k]; a.y = ha[k + 1];
                b.x = wb[k]; b.y = wb[k + 1];
                acc = __builtin_amdgcn_wmma_f32_16x16x4_f32(
                          false, a, false, b, (short)0, acc, false, false);
            }

            // sigmoid, scatter D-matrix (vgpr i -> row i + 8*(lane>>4), col n0+sub)
            #pragma unroll
            for (int i = 0; i < 8; ++i) {
                const int m = i + ((lane >> 4) << 3);
                const float s = 1.0f / (1.0f + __expf(-acc[i]));
                zt[m * ZSTR + n0 + sub] = s;
            }
        }
        __syncthreads();   // B2

        // ---- Phase 3: row L2 norms (16 threads per row) ----
        {
            const int r   = t >> 4;
            const int seg = t & 15;
            const float* zr = zt + r * ZSTR + seg * 8;
            float p = 0.f;
            #pragma unroll
            for (int i = 0; i < 8; ++i) p += zr[i] * zr[i];
            part[r][seg] = p;
        }
        __syncthreads();   // B3
        if (t < TILE_M) {
            float s = 0.f;
            #pragma unroll
            for (int i = 0; i < 16; ++i) s += part[t][i];
            normv[t] = rsqrtf(s);
        }
        __syncthreads();   // B4

        // ---- Phase 4: scale + store (two float4 per thread) ----
        {
            const int r  = t >> 4;
            const int c0 = (t & 15) * 8;
            const int gr = row0 + r;
            if (gr < B) {
                const float rn = normv[r];
                const float* zr = zt + r * ZSTR + c0;
                float4 o0, o1;
                o0.x = zr[0]*rn; o0.y = zr[1]*rn; o0.z = zr[2]*rn; o0.w = zr[3]*rn;
                o1.x = zr[4]*rn; o1.y = zr[5]*rn; o1.z = zr[6]*rn; o1.w = zr[7]*rn;
                float4* op = reinterpret_cast<float4*>(out + (size_t)gr * DIM + c0);
                op[0] = o0; op[1] = o1;
            }
        }
        // next iteration's B1 separates hl/zt reuse hazards
    }
}

extern "C" void kernel_launch(void* const* d_in, const int* in_sizes, int n_in,
                              void* d_out, int out_size, void* d_ws, size_t ws_size,
                              hipStream_t stream) {
    const int*   batch = (const int*)  d_in[0];
    const int*   neigh = (const int*)  d_in[1];
    const float* Z     = (const float*)d_in[2];
    const float* W     = (const float*)d_in[3];
    float*       out   = (float*)      d_out;

    const int B = in_sizes[0];
    const int S = (B > 0) ? (in_sizes[1] / B) : 1;
    const int nTiles = (B + TILE_M - 1) / TILE_M;
    int grid = nTiles < 1024 ? nTiles : 1024;   // persistent blocks; W staged once each

    graphsage_fused<<<grid, 256, 0, stream>>>(batch, neigh, Z, W, out, B, S, nTiles);
}